// CrystalGraphConvNet_78374563217599
// MI455X (gfx1250) — compile-verified
//
#include <hip/hip_runtime.h>
#include <math.h>

// ---------------------------------------------------------------------------
// CGCNN forward for MI455X (gfx1250, wave32, WMMA).
// Heavy op: g = total(240000x169) @ fc_w(169x169) x 3 gates x 3 layers.
// Each conv layer is ONE fused kernel: per-WG tile of 8 atoms (96 bond rows),
// `total` + all-gate `g` live in LDS (~266 KB of the 320 KB/WGP), WMMA
// V_WMMA_F32_16X16X4_F32 does the GEMM in f32 to match reference numerics.
// ---------------------------------------------------------------------------

#define N_ATOMS 20000
#define M_NBR   12
#define B_FEA   41
#define F_FEA   64
#define D_FEA   169          // 2F + B
#define ORIG    92
#define NC      3
#define KG      3            // gates per conv
#define H_FEA   128
#define N_CRYS  400
#define EPS_BN  1e-5f

#define TILE_A  8
#define R_ROWS  (TILE_A * M_NBR)   // 96 bond rows / WG
#define RT_CNT  (R_ROWS / 16)      // 6 row tiles
#define CT_CNT  11                 // ceil(169/16) col tiles
#define KSTEPS  43                 // ceil(169/4) k steps (zero padded)
#define LSTRIDE 177                // LDS row stride (odd -> fewer bank conflicts)

typedef __attribute__((ext_vector_type(2))) float v2f;
typedef __attribute__((ext_vector_type(8))) float v8f;

// ---------------------------------------------------------------------------
// Embedding: x = atom_fea(20000x92) @ emb_w(92x64) + emb_b, via f32 WMMA.
// One 16x16 output tile per wave; K loop of 23 exact 16x16x4 steps.
// ---------------------------------------------------------------------------
__global__ __launch_bounds__(256) void embed_kernel(const float* __restrict__ A,
                                                    const float* __restrict__ W,
                                                    const float* __restrict__ bias,
                                                    float* __restrict__ X) {
  const int lane = threadIdx.x & 31;
  const int wave = threadIdx.x >> 5;
  const int tile = blockIdx.x * 8 + wave;       // 5000 tiles total
  const int row0 = (tile >> 2) * 16;
  const int col0 = (tile & 3) * 16;
  const int l16  = lane & 15;
  const int hi   = lane >> 4;
  const int arow = row0 + l16;

  v8f c = {};
  for (int k0 = 0; k0 < ORIG; k0 += 4) {
    const int kk = k0 + hi * 2;                 // A/B fragment K select
    v2f a, b;
    a.x = A[arow * ORIG + kk];
    a.y = A[arow * ORIG + kk + 1];
    b.x = W[kk * F_FEA + col0 + l16];
    b.y = W[(kk + 1) * F_FEA + col0 + l16];
    c = __builtin_amdgcn_wmma_f32_16x16x4_f32(false, a, false, b, (short)0, c,
                                              false, false);
  }
  const float bv = bias[col0 + l16];
#pragma unroll
  for (int j = 0; j < 8; ++j) {
    const int r = row0 + j + 8 * hi;            // C/D layout: M = j + 8*hi
    X[r * F_FEA + col0 + l16] = c[j] + bv;
  }
}

// ---------------------------------------------------------------------------
// Fully fused conv layer.
// ---------------------------------------------------------------------------
__global__ __launch_bounds__(256) void conv_kernel(
    const float* __restrict__ x_in,    // N x 64
    const float* __restrict__ nbr_in,  // N x 12 x 41
    const int*   __restrict__ nbr_idx, // N x 12
    const float* __restrict__ fc_w,    // (3,3,169,169)
    const float* __restrict__ fc_b,    // (3,3,169)
    const float* __restrict__ bn1_g, const float* __restrict__ bn1_b,
    const float* __restrict__ bn1_m, const float* __restrict__ bn1_v,
    const float* __restrict__ bn2_g, const float* __restrict__ bn2_b,
    const float* __restrict__ bn2_m, const float* __restrict__ bn2_v,
    const float* __restrict__ atom_w, const float* __restrict__ atom_b,
    const float* __restrict__ nbr_w,  const float* __restrict__ nbr_b,
    float* __restrict__ x_out, float* __restrict__ nbr_out, int layer) {
  extern __shared__ float lds[];
  float* tot = lds;                               // R_ROWS x LSTRIDE
  float* gld = lds + R_ROWS * LSTRIDE;            // KG x R_ROWS x LSTRIDE

  const int tid   = threadIdx.x;
  const int atom0 = blockIdx.x * TILE_A;

  // ---- Phase 1: build `total` = [self(64) | gathered(64) | bond(41) | 0-pad]
  for (int i = tid; i < R_ROWS * 176; i += 256) {
    const int r = i / 176, c = i % 176;
    const int a = atom0 + r / M_NBR;
    float v = 0.f;
    if (c < F_FEA) {
      v = x_in[a * F_FEA + c];
    } else if (c < 2 * F_FEA) {
      const int nb = nbr_idx[atom0 * M_NBR + r];
      v = x_in[nb * F_FEA + (c - F_FEA)];
    } else if (c < D_FEA) {
      v = nbr_in[(atom0 * M_NBR + r) * B_FEA + (c - 2 * F_FEA)];
    }
    tot[r * LSTRIDE + c] = v;                     // cols 169..175 stay zero
  }
  __syncthreads();

  // ---- Phase 2: WMMA GEMM. Units = (gate k, col tile) distributed per wave.
  // Wave-uniform loop -> EXEC all ones for every WMMA.
  const int lane = tid & 31, wave = tid >> 5;
  const int l16 = lane & 15, hi = lane >> 4;
  for (int unit = wave; unit < KG * CT_CNT; unit += 8) {
    const int k    = unit / CT_CNT;
    const int ct   = unit % CT_CNT;
    const int col0 = ct * 16;
    const int col  = col0 + l16;
    const int colc = col < D_FEA ? col : D_FEA - 1;   // clamp (garbage unused)
    const float* W = fc_w + (size_t)((layer * KG + k) * D_FEA) * D_FEA;

    v8f acc[RT_CNT] = {};
    for (int ks = 0; ks < KSTEPS; ++ks) {
      const int kk = ks * 4 + hi * 2;
      const int d0 = kk < D_FEA ? kk : D_FEA - 1;     // clamp: A side is 0 there
      const int d1 = (kk + 1) < D_FEA ? (kk + 1) : D_FEA - 1;
      v2f b;
      b.x = W[d0 * D_FEA + colc];
      b.y = W[d1 * D_FEA + colc];
#pragma unroll
      for (int rt = 0; rt < RT_CNT; ++rt) {           // reuse B across 6 row tiles
        const int row = rt * 16 + l16;
        v2f a;
        a.x = tot[row * LSTRIDE + kk];
        a.y = tot[row * LSTRIDE + kk + 1];
        acc[rt] = __builtin_amdgcn_wmma_f32_16x16x4_f32(false, a, false, b,
                                                        (short)0, acc[rt],
                                                        false, false);
      }
    }
    // Epilogue: +fc_b, BN1 (affine with running stats), write g to LDS.
    const int pidx    = (layer * KG + k) * D_FEA + colc;
    const float scale = bn1_g[pidx] * rsqrtf(bn1_v[pidx] + EPS_BN);
    const float shift = bn1_b[pidx] - bn1_m[pidx] * scale;
    const float bias  = fc_b[pidx];
    float* gk = gld + (size_t)k * R_ROWS * LSTRIDE;
#pragma unroll
    for (int rt = 0; rt < RT_CNT; ++rt)
#pragma unroll
      for (int j = 0; j < 8; ++j) {
        const int row = rt * 16 + j + 8 * hi;
        gk[row * LSTRIDE + col0 + l16] = (acc[rt][j] + bias) * scale + shift;
      }
  }
  __syncthreads();

  // ---- Phase 3a: softmax over M, ReLU core, sum, BN2, residual, K-gate mix.
  for (int i = tid; i < TILE_A * F_FEA; i += 256) {
    const int al = i / F_FEA, f = i % F_FEA;
    const int r0 = al * M_NBR;
    float outk[KG];
#pragma unroll
    for (int k = 0; k < KG; ++k) {
      const float* gk = gld + (size_t)k * R_ROWS * LSTRIDE;
      float mx = -1e30f;
      for (int m = 0; m < M_NBR; ++m)
        mx = fmaxf(mx, gk[(r0 + m) * LSTRIDE + f]);
      float se = 0.f, s = 0.f;
      for (int m = 0; m < M_NBR; ++m) {
        const float e = __expf(gk[(r0 + m) * LSTRIDE + f] - mx);
        se += e;
        s  += e * fmaxf(gk[(r0 + m) * LSTRIDE + F_FEA + f], 0.f);
      }
      float summed = s / se;
      const int p2 = (layer * KG + k) * F_FEA + f;
      const float sc2 = bn2_g[p2] * rsqrtf(bn2_v[p2] + EPS_BN);
      const float sh2 = bn2_b[p2] - bn2_m[p2] * sc2;
      summed = summed * sc2 + sh2;
      outk[k] = x_in[(atom0 + al) * F_FEA + f] + summed;
    }
    float og[2 * KG];
#pragma unroll
    for (int j = 0; j < 2 * KG; ++j) {
      float s = atom_b[layer * 2 * KG + j];
#pragma unroll
      for (int k = 0; k < KG; ++k)
        s += outk[k] * atom_w[(layer * KG + k) * 2 * KG + j];
      og[j] = s;
    }
    const float m3 = fmaxf(fmaxf(og[3], og[4]), og[5]);
    const float e0 = __expf(og[3] - m3), e1 = __expf(og[4] - m3),
                e2 = __expf(og[5] - m3);
    const float inv = 1.f / (e0 + e1 + e2);
    x_out[(atom0 + al) * F_FEA + f] = (og[0] * e0 + og[1] * e1 + og[2] * e2) * inv;
  }

  // ---- Phase 3b: bond residual + K-gate mix.
  for (int i = tid; i < R_ROWS * B_FEA; i += 256) {
    const int r = i / B_FEA, b = i % B_FEA;
    const float nin = nbr_in[(atom0 * M_NBR + r) * B_FEA + b];
    float nk[KG];
#pragma unroll
    for (int k = 0; k < KG; ++k)
      nk[k] = gld[((size_t)k * R_ROWS + r) * LSTRIDE + 2 * F_FEA + b] + nin;
    float ng[2 * KG];
#pragma unroll
    for (int j = 0; j < 2 * KG; ++j) {
      float s = nbr_b[layer * 2 * KG + j];
#pragma unroll
      for (int k = 0; k < KG; ++k)
        s += nk[k] * nbr_w[(layer * KG + k) * 2 * KG + j];
      ng[j] = s;
    }
    const float m3 = fmaxf(fmaxf(ng[3], ng[4]), ng[5]);
    const float e0 = __expf(ng[3] - m3), e1 = __expf(ng[4] - m3),
                e2 = __expf(ng[5] - m3);
    const float inv = 1.f / (e0 + e1 + e2);
    nbr_out[(atom0 * M_NBR + r) * B_FEA + b] =
        (ng[0] * e0 + ng[1] * e1 + ng[2] * e2) * inv;
  }
}

// ---------------------------------------------------------------------------
// Pooling + head.
// ---------------------------------------------------------------------------
__global__ void zero_kernel(float* sums, float* counts) {
  const int i = blockIdx.x * blockDim.x + threadIdx.x;
  if (i < N_CRYS * F_FEA) sums[i] = 0.f;
  if (i < N_CRYS) counts[i] = 0.f;
}

__global__ void scatter_kernel(const float* __restrict__ x,
                               const int* __restrict__ cidx,
                               float* sums, float* counts) {
  const int i = blockIdx.x * blockDim.x + threadIdx.x;
  if (i >= N_ATOMS * F_FEA) return;
  const int n = i / F_FEA, f = i % F_FEA;
  const int c = cidx[n];
  atomicAdd(&sums[c * F_FEA + f], x[i]);     // global_atomic_add_f32
  if (f == 0) atomicAdd(&counts[c], 1.f);
}

__global__ __launch_bounds__(128) void head_kernel(
    const float* __restrict__ sums, const float* __restrict__ counts,
    const float* __restrict__ fc1_w, const float* __restrict__ fc1_b,
    const float* __restrict__ out_w, const float* __restrict__ out_b,
    float* __restrict__ out) {
  __shared__ float meanf[F_FEA];
  __shared__ float red[H_FEA];
  const int c = blockIdx.x, h = threadIdx.x;
  const float cnt = fmaxf(counts[c], 1.f);
  if (h < F_FEA) meanf[h] = fmaxf(sums[c * F_FEA + h] / cnt, 0.f);  // relu(mean)
  __syncthreads();
  float s = fc1_b[h];
  for (int f = 0; f < F_FEA; ++f) s += meanf[f] * fc1_w[f * H_FEA + h];
  red[h] = fmaxf(s, 0.f) * out_w[h];
  __syncthreads();
  for (int off = 64; off > 0; off >>= 1) {
    if (h < off) red[h] += red[h + off];
    __syncthreads();
  }
  if (h == 0) out[c] = red[0] + out_b[0];
}

// ---------------------------------------------------------------------------
extern "C" void kernel_launch(void* const* d_in, const int* in_sizes, int n_in,
                              void* d_out, int out_size, void* d_ws,
                              size_t ws_size, hipStream_t stream) {
  const float* atom_fea = (const float*)d_in[0];
  const float* nbr_fea  = (const float*)d_in[1];
  const int*   nbr_idx  = (const int*)d_in[2];
  const int*   cidx     = (const int*)d_in[3];
  // d_in[4] = n_crystals (fixed 400 per reference constants)
  const float* emb_w = (const float*)d_in[5];
  const float* emb_b = (const float*)d_in[6];
  const float* fc_w  = (const float*)d_in[7];
  const float* fc_b  = (const float*)d_in[8];
  const float* bn1_g = (const float*)d_in[9];
  const float* bn1_b = (const float*)d_in[10];
  const float* bn1_m = (const float*)d_in[11];
  const float* bn1_v = (const float*)d_in[12];
  const float* bn2_g = (const float*)d_in[13];
  const float* bn2_b = (const float*)d_in[14];
  const float* bn2_m = (const float*)d_in[15];
  const float* bn2_v = (const float*)d_in[16];
  const float* atom_w = (const float*)d_in[17];
  const float* atom_b = (const float*)d_in[18];
  const float* nbr_w  = (const float*)d_in[19];
  const float* nbr_b  = (const float*)d_in[20];
  const float* fc1_w = (const float*)d_in[21];
  const float* fc1_b = (const float*)d_in[22];
  const float* out_w = (const float*)d_in[23];
  const float* out_b = (const float*)d_in[24];

  // Workspace layout (~89 MB): x ping-pong (2x5.1MB) + nbr ping-pong (2x39MB)
  // + pooling sums/counts.
  float* ws = (float*)d_ws;
  size_t off = 0;
  float* x0  = ws + off; off += (size_t)N_ATOMS * F_FEA;
  float* x1  = ws + off; off += (size_t)N_ATOMS * F_FEA;
  float* nb0 = ws + off; off += (size_t)N_ATOMS * M_NBR * B_FEA;
  float* nb1 = ws + off; off += (size_t)N_ATOMS * M_NBR * B_FEA;
  float* sums   = ws + off; off += (size_t)N_CRYS * F_FEA;
  float* counts = ws + off;

  const size_t ldsBytes =
      (size_t)(1 + KG) * R_ROWS * LSTRIDE * sizeof(float);  // ~266 KB
  (void)hipFuncSetAttribute((const void*)conv_kernel,
                            hipFuncAttributeMaxDynamicSharedMemorySize,
                            (int)ldsBytes);

  // Embedding
  embed_kernel<<<(N_ATOMS / 16) * (F_FEA / 16) / 8, 256, 0, stream>>>(
      atom_fea, emb_w, emb_b, x0);

  // Three fused conv layers (ping-pong through workspace)
  const int nBlocks = N_ATOMS / TILE_A;  // 2500
  conv_kernel<<<nBlocks, 256, ldsBytes, stream>>>(
      x0, nbr_fea, nbr_idx, fc_w, fc_b, bn1_g, bn1_b, bn1_m, bn1_v,
      bn2_g, bn2_b, bn2_m, bn2_v, atom_w, atom_b, nbr_w, nbr_b, x1, nb0, 0);
  conv_kernel<<<nBlocks, 256, ldsBytes, stream>>>(
      x1, nb0, nbr_idx, fc_w, fc_b, bn1_g, bn1_b, bn1_m, bn1_v,
      bn2_g, bn2_b, bn2_m, bn2_v, atom_w, atom_b, nbr_w, nbr_b, x0, nb1, 1);
  conv_kernel<<<nBlocks, 256, ldsBytes, stream>>>(
      x0, nb1, nbr_idx, fc_w, fc_b, bn1_g, bn1_b, bn1_m, bn1_v,
      bn2_g, bn2_b, bn2_m, bn2_v, atom_w, atom_b, nbr_w, nbr_b, x1, nb0, 2);

  // Per-crystal mean pooling + 64->128->1 head
  zero_kernel<<<(N_CRYS * F_FEA + 255) / 256, 256, 0, stream>>>(sums, counts);
  scatter_kernel<<<(N_ATOMS * F_FEA + 255) / 256, 256, 0, stream>>>(
      x1, cidx, sums, counts);
  head_kernel<<<N_CRYS, H_FEA, 0, stream>>>(sums, counts, fc1_w, fc1_b, out_w,
                                            out_b, (float*)d_out);
}